// MemNet_66151086293669
// MI455X (gfx1250) — compile-verified
//
#include <hip/hip_runtime.h>
#include <hip/hip_bf16.h>

typedef float v2f __attribute__((ext_vector_type(2)));
typedef float v8f __attribute__((ext_vector_type(8)));

#define EPS_MASK 1e-7f
#define EPS_NORM 1e-12f
#define EPS_COS  1e-8f

// ---------------------------------------------------------------- utilities
__global__ __launch_bounds__(256) void zero_kernel(float* p, int n) {
    int i = blockIdx.x * 256 + threadIdx.x;
    if (i < n) p[i] = 0.0f;
}

// ---------------------------------------------------------------- mask: 4x4 avg pool -> threshold -> eps floor
__global__ __launch_bounds__(256) void mask_kernel(const float* __restrict__ tm,
                                                   float* __restrict__ m, int N) {
    int n = blockIdx.x * 256 + threadIdx.x;
    if (n >= N) return;
    int b = n >> 12, hw = n & 4095, h = hw >> 6, w = hw & 63;
    const float* base = tm + (size_t)b * 65536 + (size_t)(h * 4) * 256 + w * 4;
    float s = 0.0f;
    for (int r = 0; r < 4; ++r) {
        float4 v = *reinterpret_cast<const float4*>(base + r * 256);
        s += v.x + v.y + v.z + v.w;
    }
    s *= (1.0f / 16.0f);
    m[n] = (s > 0.5f) ? 1.0f : EPS_MASK;
}

// ---------------------------------------------------------------- normalize: feat*m -> query (NCHW out) + qp [N,256] + |query|
__global__ __launch_bounds__(256) void norm_kernel(const float* __restrict__ feat,
                                                   const float* __restrict__ m,
                                                   float* __restrict__ qp,
                                                   float* __restrict__ nrm2,
                                                   float* __restrict__ out_q) {
    __shared__ float ls[256][65];
    __shared__ float red[4][64];
    __shared__ float m_ls[64];
    __shared__ float s_qs[64];
    __shared__ float s_ps[64];
    int t = threadIdx.x;
    int n0 = blockIdx.x * 64;
    int b = n0 >> 12;
    int hw0 = n0 & 4095;
    if (t < 64) m_ls[t] = m[n0 + t];
    __syncthreads();
    int nl = t & 63;
    int cq = t >> 6;
    for (int i = 0; i < 64; ++i) {
        int c = (i << 2) + cq;
        ls[c][nl] = feat[((size_t)(b * 256 + c) << 12) + hw0 + nl] * m_ls[nl];
    }
    __syncthreads();
    float ss = 0.0f;
    for (int j = 0; j < 64; ++j) {
        float v = ls[cq * 64 + j][nl];
        ss += v * v;
    }
    red[cq][nl] = ss;
    __syncthreads();
    if (t < 64) {
        float s = red[0][t] + red[1][t] + red[2][t] + red[3][t];
        float n1 = sqrtf(s);
        float s1 = 1.0f / fmaxf(n1, EPS_NORM);
        float nr = n1 * s1;                      // |query| after first normalize
        float s2 = 1.0f / fmaxf(nr, EPS_NORM);   // faithful second normalize
        nrm2[n0 + t] = nr;
        s_qs[t] = s1;
        s_ps[t] = s1 * s2;
    }
    __syncthreads();
    for (int i = 0; i < 64; ++i)
        qp[(size_t)(n0 + i) * 256 + t] = ls[t][i] * s_ps[i];
    for (int i = 0; i < 64; ++i) {
        int c = (i << 2) + cq;
        out_q[((size_t)(b * 256 + c) << 12) + hw0 + nl] = ls[c][nl] * s_qs[nl];
    }
}

// ---------------------------------------------------------------- WMMA GEMM with async global->LDS double buffering
// C[n,m] = sum_k A[n,k] * (BTRANS ? B[m,k] : B[k,m])
// Block tile 128x128, K-chunk 16, 8 waves (4 row-blocks x 2 col-blocks),
// each wave: 2x4 tiles of v_wmma_f32_16x16x4_f32.
// Tiles staged via GLOBAL_LOAD_ASYNC_TO_LDS_B128 (ASYNCcnt), ping-pong buffers.
// Row strides padded to 16B multiples so every async LDS destination is aligned.
template <bool BTRANS>
__global__ __launch_bounds__(256) void gemm_core(const float* __restrict__ A,
                                                 const float* __restrict__ B,
                                                 float* __restrict__ C,
                                                 int K, int lda, int ldb, int ldc) {
    constexpr int BR = BTRANS ? 128 : 16;
    constexpr int BS = BTRANS ? 20 : 132;   // 80B / 528B row strides (16B multiples)
    __shared__ float lsA[2][128][20];
    __shared__ float lsB[2][BR][BS];
    int t = threadIdx.x;
    int lane = t & 31;
    int wave = t >> 5;
    int wr = wave & 3;       // 32-row block
    int wc = wave >> 2;      // 64-col block
    int n0 = blockIdx.x * 128;
    int m0 = blockIdx.y * 128;
    int lhalf = lane >> 4;
    int lmod = lane & 15;

    int rA = t >> 2;                 // 0..63 (two passes of 64 rows)
    int k4 = (t & 3) << 2;           // 0,4,8,12
    int kl = t >> 5;                 // 0..7 (B k-major loader)
    int c4 = (lane) << 2;            // 0..124

    v8f zero = {};
    v8f acc[2][4];
    for (int i = 0; i < 2; ++i)
        for (int j = 0; j < 4; ++j) acc[i][j] = zero;

    // issue one K-chunk's tiles (4 async b128 per thread)
    auto issue = [&](int buf, int kc) {
        for (int p = 0; p < 2; ++p) {
            int row = rA + p * 64;
            const float* ga = A + (size_t)(n0 + row) * lda + kc + k4;
            unsigned la = (unsigned)(uintptr_t)&lsA[buf][row][k4];
            asm volatile("global_load_async_to_lds_b128 %0, %1, off"
                         :: "v"(la), "v"(ga) : "memory");
        }
        if constexpr (BTRANS) {
            // m-major tile: straight row copy; fragments read lsB[col][k..k+1]
            for (int p = 0; p < 2; ++p) {
                int mm = rA + p * 64;
                const float* gb = B + (size_t)(m0 + mm) * ldb + kc + k4;
                unsigned lb = (unsigned)(uintptr_t)&lsB[buf][mm][k4];
                asm volatile("global_load_async_to_lds_b128 %0, %1, off"
                             :: "v"(lb), "v"(gb) : "memory");
            }
        } else {
            // k-major tile: B rows are contiguous; straight row copy
            for (int p = 0; p < 2; ++p) {
                int kk2 = kl + p * 8;
                const float* gb = B + (size_t)(kc + kk2) * ldb + m0 + c4;
                unsigned lb = (unsigned)(uintptr_t)&lsB[buf][kk2][c4];
                asm volatile("global_load_async_to_lds_b128 %0, %1, off"
                             :: "v"(lb), "v"(gb) : "memory");
            }
        }
    };

    int nch = K >> 4;
    issue(0, 0);
    for (int i = 0; i < nch; ++i) {
        int cur = i & 1;
        if (i + 1 < nch) {
            issue(cur ^ 1, (i + 1) << 4);
            // 8 outstanding, in-order completion: <=4 left means chunk i landed
            asm volatile("s_wait_asynccnt 0x4" ::: "memory");
        } else {
            asm volatile("s_wait_asynccnt 0x0" ::: "memory");
        }
        __syncthreads();   // make all waves' async copies visible
#pragma unroll
        for (int kk = 0; kk < 16; kk += 4) {
            int ko = kk + lhalf * 2;  // lanes 0-15 -> K{0,1}; lanes 16-31 -> K{2,3}
            v2f a[2], bb[4];
#pragma unroll
            for (int rt = 0; rt < 2; ++rt) {
                int row = wr * 32 + rt * 16 + lmod;
                a[rt] = *reinterpret_cast<const v2f*>(&lsA[cur][row][ko]);
            }
#pragma unroll
            for (int ct = 0; ct < 4; ++ct) {
                int col = wc * 64 + ct * 16 + lmod;
                if constexpr (BTRANS) {
                    bb[ct] = *reinterpret_cast<const v2f*>(&lsB[cur][col][ko]);
                } else {
                    bb[ct].x = lsB[cur][ko][col];
                    bb[ct].y = lsB[cur][ko + 1][col];
                }
            }
#pragma unroll
            for (int rt = 0; rt < 2; ++rt)
#pragma unroll
                for (int ct = 0; ct < 4; ++ct)
                    acc[rt][ct] = __builtin_amdgcn_wmma_f32_16x16x4_f32(
                        false, a[rt], false, bb[ct], (short)0, acc[rt][ct], false, false);
        }
        __syncthreads();   // readers done before buffer is overwritten next iter
    }
    // epilogue: C/D layout — VGPR i: M = (lane>=16 ? 8:0)+i, N = lane%16
    for (int rt = 0; rt < 2; ++rt) {
        int rbase = n0 + wr * 32 + rt * 16 + lhalf * 8;
        for (int ct = 0; ct < 4; ++ct) {
            int col = m0 + wc * 64 + ct * 16 + lmod;
            for (int i = 0; i < 8; ++i)
                C[(size_t)(rbase + i) * ldc + col] = acc[rt][ct][i];
        }
    }
}

// ---------------------------------------------------------------- column max (softmax-over-tokens collapses to this)
__global__ __launch_bounds__(256) void colmax_part(const float* __restrict__ score,
                                                   float* __restrict__ part) {
    int t = threadIdx.x;
    int blk = blockIdx.x;
    float m0 = -3.4e38f, m1 = -3.4e38f;
    for (int r = 0; r < 256; ++r) {
        size_t row = (size_t)(blk * 256 + r) * 512;
        m0 = fmaxf(m0, score[row + t]);
        m1 = fmaxf(m1, score[row + 256 + t]);
    }
    part[(size_t)blk * 512 + t] = m0;
    part[(size_t)blk * 512 + 256 + t] = m1;
}

__global__ __launch_bounds__(256) void colmax_final(const float* __restrict__ part,
                                                    float* __restrict__ colmax) {
    int col = blockIdx.x * 256 + threadIdx.x;
    float mv = -3.4e38f;
    for (int b = 0; b < 256; ++b) mv = fmaxf(mv, part[(size_t)b * 512 + col]);
    colmax[col] = mv;
}

// ---------------------------------------------------------------- row argmax + wgt = exp(rowmax - colmax[g]) + scatter-add
__global__ __launch_bounds__(256) void row_update(const float* __restrict__ score,
                                                  const float* __restrict__ colmax,
                                                  const float* __restrict__ qp,
                                                  float* __restrict__ qu) {
    __shared__ float sval[256];
    __shared__ int sidx[256];
    __shared__ float swgt;
    __shared__ int sg;
    int t = threadIdx.x;
    int n = blockIdx.x;
    float v0 = score[(size_t)n * 512 + t];
    float v1 = score[(size_t)n * 512 + 256 + t];
    float bv; int bi;
    if (v1 > v0) { bv = v1; bi = t + 256; } else { bv = v0; bi = t; }
    sval[t] = bv; sidx[t] = bi;
    __syncthreads();
    for (int s = 128; s > 0; s >>= 1) {
        if (t < s) {
            float ov = sval[t + s]; int oi = sidx[t + s];
            if (ov > sval[t] || (ov == sval[t] && oi < sidx[t])) { sval[t] = ov; sidx[t] = oi; }
        }
        __syncthreads();
    }
    if (t == 0) {
        sg = sidx[0];
        swgt = __expf(sval[0] - colmax[sidx[0]]);
    }
    __syncthreads();
    atomicAdd(&qu[(size_t)sg * 256 + t], swgt * qp[(size_t)n * 256 + t]);
}

// ---------------------------------------------------------------- updated_memory = l2norm(qu + mem)
__global__ __launch_bounds__(256) void update_mem(const float* __restrict__ qu,
                                                  const float* __restrict__ mem,
                                                  float* __restrict__ UM,
                                                  float* __restrict__ out_um) {
    __shared__ float red[256];
    int t = threadIdx.x;
    int mr = blockIdx.x;
    float v = qu[(size_t)mr * 256 + t] + mem[(size_t)mr * 256 + t];
    red[t] = v * v;
    __syncthreads();
    for (int s = 128; s > 0; s >>= 1) {
        if (t < s) red[t] += red[t + s];
        __syncthreads();
    }
    float s1 = 1.0f / fmaxf(sqrtf(red[0]), EPS_NORM);
    float um = v * s1;
    UM[(size_t)mr * 256 + t] = um;
    out_um[(size_t)mr * 256 + t] = um;
}

// ---------------------------------------------------------------- in-place row softmax over 512 slots
__global__ __launch_bounds__(256) void row_softmax(float* __restrict__ score) {
    __shared__ float red[256];
    int t = threadIdx.x;
    size_t base = (size_t)blockIdx.x * 512;
    float v0 = score[base + t], v1 = score[base + 256 + t];
    red[t] = fmaxf(v0, v1);
    __syncthreads();
    for (int s = 128; s > 0; s >>= 1) {
        if (t < s) red[t] = fmaxf(red[t], red[t + s]);
        __syncthreads();
    }
    float mx = red[0];
    __syncthreads();
    float e0 = __expf(v0 - mx), e1 = __expf(v1 - mx);
    red[t] = e0 + e1;
    __syncthreads();
    for (int s = 128; s > 0; s >>= 1) {
        if (t < s) red[t] += red[t + s];
        __syncthreads();
    }
    float inv = 1.0f / red[0];
    score[base + t] = e0 * inv;
    score[base + 256 + t] = e1 * inv;
}

// ---------------------------------------------------------------- cosine sim + updated_query output
__global__ __launch_bounds__(256) void sim_out(const float* __restrict__ qp,
                                               const float* __restrict__ att2,
                                               const float* __restrict__ nrm2,
                                               float* __restrict__ out_uq) {
    __shared__ float rd[256], ra[256];
    __shared__ float ssc;
    int t = threadIdx.x;
    int n = blockIdx.x;
    float q = qp[(size_t)n * 256 + t];
    float a = att2[(size_t)n * 256 + t];
    rd[t] = q * a; ra[t] = a * a;
    __syncthreads();
    for (int s = 128; s > 0; s >>= 1) {
        if (t < s) { rd[t] += rd[t + s]; ra[t] += ra[t + s]; }
        __syncthreads();
    }
    if (t == 0) {
        float nr = nrm2[n];
        float qs = fmaxf(nr, EPS_NORM);         // query = qp * qs
        float num = qs * rd[0];
        float den = fmaxf(nr * sqrtf(ra[0]), EPS_COS);
        ssc = (num / den) * qs;                 // sim*qs: uq = qp * (sim*qs)
    }
    __syncthreads();
    int b = n >> 12, hw = n & 4095;
    out_uq[((size_t)(b * 256 + t) << 12) + hw] = q * ssc;
}

// ---------------------------------------------------------------- launch
extern "C" void kernel_launch(void* const* d_in, const int* in_sizes, int n_in,
                              void* d_out, int out_size, void* d_ws, size_t ws_size,
                              hipStream_t stream) {
    const float* feat  = (const float*)d_in[0];   // [16,256,64,64]
    const float* tmask = (const float*)d_in[1];   // [16,256,256]
    const float* mem   = (const float*)d_in[2];   // [512,256]

    const int N = 65536;  // 16*64*64 tokens
    float* out    = (float*)d_out;
    float* out_uq = out;
    float* out_q  = out + (size_t)16777216;
    float* out_um = out + (size_t)33554432;

    float* ws    = (float*)d_ws;
    float* qp    = ws;                       // N*256
    float* score = qp + (size_t)16777216;    // N*512 (reused for score2/sm2)
    float* att2  = score + (size_t)33554432; // N*256
    float* mbuf  = att2 + (size_t)16777216;  // N
    float* nrm2  = mbuf + 65536;             // N
    float* cmax  = nrm2 + 65536;             // 512
    float* part  = cmax + 512;               // 256*512
    float* qu    = part + 131072;            // 512*256
    float* UM    = qu + 131072;              // 512*256

    zero_kernel<<<512, 256, 0, stream>>>(qu, 131072);
    mask_kernel<<<N / 256, 256, 0, stream>>>(tmask, mbuf, N);
    norm_kernel<<<N / 64, 256, 0, stream>>>(feat, mbuf, qp, nrm2, out_q);

    // score = qp @ mem.T  (WMMA + async LDS staging)
    gemm_core<true><<<dim3(N / 128, 4), 256, 0, stream>>>(qp, mem, score, 256, 256, 256, 512);
    colmax_part<<<256, 256, 0, stream>>>(score, part);
    colmax_final<<<2, 256, 0, stream>>>(part, cmax);
    row_update<<<N, 256, 0, stream>>>(score, cmax, qp, qu);
    update_mem<<<512, 256, 0, stream>>>(qu, mem, UM, out_um);

    // score2 = qp @ UM.T  (WMMA), softmax rows -> sm2 (in place)
    gemm_core<true><<<dim3(N / 128, 4), 256, 0, stream>>>(qp, UM, score, 256, 256, 256, 512);
    row_softmax<<<N, 256, 0, stream>>>(score);

    // att2 = sm2 @ UM  (WMMA)
    gemm_core<false><<<dim3(N / 128, 2), 256, 0, stream>>>(score, UM, att2, 512, 512, 256, 256);
    sim_out<<<N, 256, 0, stream>>>(qp, att2, nrm2, out_uq);
}